// ReLUUpsample2xWithInterpolation_69011534512367
// MI455X (gfx1250) — compile-verified
//
#include <hip/hip_runtime.h>

typedef __attribute__((ext_vector_type(2))) float v2f;
typedef __attribute__((ext_vector_type(8))) float v8f;

#define LRELU_SLOPE 0.01f
#define CUBIC_A    (-0.75f)

// ---------------------------------------------------------------------------
// Pass 1: channel contraction z[b,o,p] = sum_c W[o,c] * x[b,c,p]
// per batch: M=128 (C_OUT) x N=16384 (pixels) x K=256 (C_IN), fp32
// Block = 256 threads = 8 waves. Block handles one (batch, 16-pixel) N-tile,
// wave w handles M-tile w. x tile staged in LDS once (x streamed from HBM 1x).
// ---------------------------------------------------------------------------
__global__ __launch_bounds__(256) void gemm_channels_wmma(
    const float* __restrict__ x,   // [8][256][16384]
    const float* __restrict__ w,   // [128][256]
    float* __restrict__ z)         // [8][128][16384]
{
    __shared__ float ldsB[256 * 16];   // 16 KB: [c][p] tile

    const int t     = threadIdx.x;
    const int b     = blockIdx.x >> 10;     // / 1024 N-tiles per batch
    const int ptile = blockIdx.x & 1023;
    const int p0    = ptile << 4;

    // Stage x[b][c][p0..p0+15] for c = threadIdx.x (256 threads = 256 channels)
    {
        const float4* src = (const float4*)(x + (((size_t)(b * 256 + t)) << 14) + p0);
        float4* dst = (float4*)(ldsB + t * 16);
        dst[0] = src[0];
        dst[1] = src[1];
        dst[2] = src[2];
        dst[3] = src[3];
    }
    __syncthreads();

    const int lane  = t & 31;
    const int mwave = t >> 5;        // M-tile 0..7
    const int n     = lane & 15;     // pixel within tile / M-row for A
    const int khalf = lane >> 4;     // 0: K rows k,k+1 ; 1: K rows k+2,k+3

    // A-fragment source: weight row m = mwave*16 + (lane&15), contiguous K pair
    const float* wrow = w + (mwave * 16 + n) * 256 + (khalf << 1);

    v8f acc = {};
#pragma unroll 8
    for (int k = 0; k < 256; k += 4) {
        v2f afrag;                       // ISA 16x4 A layout: vgpr0=K{0|2}, vgpr1=K{1|3}
        afrag.x = wrow[k];
        afrag.y = wrow[k + 1];

        const int kb = k + (khalf << 1); // ISA 4x16 B layout mirrors A K-split
        v2f bfrag;
        bfrag.x = ldsB[kb * 16 + n];
        bfrag.y = ldsB[(kb + 1) * 16 + n];

        acc = __builtin_amdgcn_wmma_f32_16x16x4_f32(
            /*neg_a=*/false, afrag, /*neg_b=*/false, bfrag,
            /*c_mod=*/(short)0, acc, /*reuse_a=*/false, /*reuse_b=*/false);
    }

    // C/D layout: vgpr r holds M=r (lanes 0-15) and M=r+8 (lanes 16-31), N=lane&15
    float* zp = z + (((size_t)(b * 128 + (mwave << 4) + (khalf << 3))) << 14) + p0 + n;
#pragma unroll
    for (int r = 0; r < 8; ++r) {
        zp[(size_t)r << 14] = acc[r];
    }
}

// ---------------------------------------------------------------------------
// Pass 2: bicubic 2x upsample (align_corners=True, border clamp) of z,
// + bias + LeakyReLU.  One thread per output pixel, direct 4x4 gather.
// ---------------------------------------------------------------------------
__device__ __forceinline__ float cub_near(float s) {   // 0 <= s <= 1
    return ((CUBIC_A + 2.0f) * s - (CUBIC_A + 3.0f)) * s * s + 1.0f;
}
__device__ __forceinline__ float cub_far(float s) {    // 1 <= s <= 2
    return (((s - 5.0f) * s + 8.0f) * s - 4.0f) * CUBIC_A;
}

__global__ __launch_bounds__(256) void upsample_bias_lrelu(
    const float* __restrict__ z,     // [8][128][128][128]
    const float* __restrict__ bias,  // [128]
    float* __restrict__ out)         // [8][128][256][256]
{
    const size_t idx = (size_t)blockIdx.x * 256 + threadIdx.x;
    const int j = (int)(idx & 255);
    const int i = (int)((idx >> 8) & 255);
    const int o = (int)((idx >> 16) & 127);
    const int b = (int)(idx >> 23);

    const float scale = 127.0f / 255.0f;   // (H-1)/(2H-1), align_corners=True

    float fy = (float)i * scale;
    int   iy = (int)fy;
    float ty = fy - (float)iy;
    float fx = (float)j * scale;
    int   ix = (int)fx;
    float tx = fx - (float)ix;

    float wy[4] = { cub_far(ty + 1.0f), cub_near(ty), cub_near(1.0f - ty), cub_far(2.0f - ty) };
    float wx[4] = { cub_far(tx + 1.0f), cub_near(tx), cub_near(1.0f - tx), cub_far(2.0f - tx) };

    int ys[4] = { max(iy - 1, 0), iy, min(iy + 1, 127), min(iy + 2, 127) };
    int xs[4] = { max(ix - 1, 0), ix, min(ix + 1, 127), min(ix + 2, 127) };

    const float* zp = z + (((size_t)(b * 128 + o)) << 14);

    float s = 0.0f;
#pragma unroll
    for (int ky = 0; ky < 4; ++ky) {
        const float* row = zp + (ys[ky] << 7);
        float r = wx[0] * row[xs[0]] + wx[1] * row[xs[1]]
                + wx[2] * row[xs[2]] + wx[3] * row[xs[3]];
        s += wy[ky] * r;
    }

    float v = s + bias[o];
    out[idx] = (v >= 0.0f) ? v : LRELU_SLOPE * v;
}

// ---------------------------------------------------------------------------
extern "C" void kernel_launch(void* const* d_in, const int* in_sizes, int n_in,
                              void* d_out, int out_size, void* d_ws, size_t ws_size,
                              hipStream_t stream) {
    const float* x    = (const float*)d_in[0];  // [8][256][128][128]
    const float* w    = (const float*)d_in[1];  // [128][256]
    const float* bias = (const float*)d_in[2];  // [128]
    float* out = (float*)d_out;                 // [8][128][256][256]
    float* z   = (float*)d_ws;                  // needs 64 MB scratch

    // Pass 1: 8 batches * 1024 N-tiles = 8192 blocks
    gemm_channels_wmma<<<8192, 256, 0, stream>>>(x, w, z);

    // Pass 2: 67,108,864 outputs / 256 = 262,144 blocks
    upsample_bias_lrelu<<<262144, 256, 0, stream>>>(z, bias, out);
}